// Yolact_31018253811957
// MI455X (gfx1250) — compile-verified
//
#include <hip/hip_runtime.h>
#include <hip/hip_bf16.h>
#include <cstdint>

#define AS1 __attribute__((address_space(1)))
#define AS3 __attribute__((address_space(3)))

typedef float v2f __attribute__((ext_vector_type(2)));
typedef float v8f __attribute__((ext_vector_type(8)));
typedef int   v4i __attribute__((ext_vector_type(4)));

namespace {
constexpr int kB       = 16;
constexpr int kN       = 19248;   // priors
constexpr int kC       = 80;      // classes
constexpr int kK       = 200;     // top-k
constexpr int kKP      = 208;     // 13*16, WMMA-padded
constexpr int kNT      = 13;      // 16x16 tiles along K
constexpr int kPairs   = kNT * (kNT + 1) / 2;  // 91 upper-tri tile pairs
constexpr int kThreads = 256;     // 8 waves (wave32)
constexpr int kCand    = 512;     // candidate/sort buffer (pow2)
constexpr float kIouThr  = 0.5f;
constexpr float kConfThr = 0.05f;
}

__device__ __forceinline__ void wait_async_zero() {
#if defined(__HIP_DEVICE_COMPILE__) && __has_builtin(__builtin_amdgcn_s_wait_asynccnt)
  __builtin_amdgcn_s_wait_asynccnt(0);
#elif defined(__HIP_DEVICE_COMPILE__)
  asm volatile("s_wait_asynccnt 0" ::: "memory");
#endif
}

__global__ __launch_bounds__(kThreads)
void yolact_fastnms_fused(const float* __restrict__ boxes_raw,  // [B,N,4] cx,cy,w,h
                          const float* __restrict__ scores,     // [B,C,N]
                          float* __restrict__ out) {            // [B,C,K,5]
  extern __shared__ __align__(16) unsigned int skey[];          // kN score keys
  __shared__ unsigned int hist[256];
  __shared__ unsigned long long cand[kCand];
  __shared__ unsigned int sel_prefix, sel_rank, cand_count;
  __shared__ float bx1[kKP], by1[kKP], bx2[kKP], by2[kKP], barea[kKP];
  __shared__ unsigned int sup[kKP];   // per-column suppression flag

  const unsigned t  = threadIdx.x;
  const unsigned bc = blockIdx.x;              // b*C + c
  const unsigned b  = bc / kC;
  const float* row  = scores + (size_t)bc * kN;

  // ---------- Phase 1: stage score row -> LDS (CDNA5 async DMA, ASYNCcnt) ----------
#if defined(__HIP_DEVICE_COMPILE__) && __has_builtin(__builtin_amdgcn_global_load_async_to_lds_b128)
  for (unsigned u = t; u < (unsigned)(kN / 4); u += kThreads) {
    __builtin_amdgcn_global_load_async_to_lds_b128(
        (AS1 v4i*)(void*)(const_cast<float*>(row) + 4u * u),
        (AS3 v4i*)(void*)((char*)skey + 16u * u),
        0, 0);
  }
  wait_async_zero();
#else
  {
    const uint4* row4 = reinterpret_cast<const uint4*>(row);
    uint4* s4 = reinterpret_cast<uint4*>(skey);
    for (unsigned u = t; u < (unsigned)(kN / 4); u += kThreads) s4[u] = row4[u];
  }
#endif
  if (t == 0) { sel_prefix = 0u; sel_rank = (unsigned)kK; }
  __syncthreads();

  // ---------- Phase 2: exact 4-pass MSB radix select (200th largest) ----------
  // Keys are float bits; scores >= 0 so uint order == float order.
  for (int pass = 0; pass < 4; ++pass) {
    const unsigned pfx = sel_prefix;
    hist[t] = 0u;
    __syncthreads();
    const int shift = 24 - 8 * pass;
    const unsigned maskHi = pass ? (0xFFFFFFFFu << (shift + 8)) : 0u;
    for (unsigned u = t; u < (unsigned)kN; u += kThreads) {
      const unsigned k = skey[u];
      if ((k & maskHi) == pfx) atomicAdd(&hist[(k >> shift) & 255u], 1u);
    }
    __syncthreads();
    if (t == 0) {
      unsigned r = sel_rank, cum = 0u;
      for (int d = 255; d >= 0; --d) {
        const unsigned c = hist[d];
        if (cum + c >= r) { sel_prefix = pfx | ((unsigned)d << shift); sel_rank = r - cum; break; }
        cum += c;
      }
    }
    __syncthreads();
  }
  const unsigned T = sel_prefix;  // exact 200th-largest key

  // ---------- Phase 3: compact candidates (key >= T) ----------
  if (t == 0) cand_count = 0u;
  __syncthreads();
  for (unsigned u = t; u < (unsigned)kN; u += kThreads) {
    const unsigned k = skey[u];
    if (k >= T) {
      const unsigned pos = atomicAdd(&cand_count, 1u);
      if (pos < (unsigned)kCand)
        cand[pos] = ((unsigned long long)k << 32) | (unsigned)(~u);  // score desc, idx asc
    }
  }
  __syncthreads();
  const unsigned M = (cand_count < (unsigned)kCand) ? cand_count : (unsigned)kCand;
  for (unsigned p = t; p < (unsigned)kCand; p += kThreads)
    if (p >= M) cand[p] = 0ull;

  // ---------- Phase 4: bitonic sort 512 (descending) ----------
  for (unsigned kk = 2; kk <= (unsigned)kCand; kk <<= 1) {
    for (unsigned jj = kk >> 1; jj > 0; jj >>= 1) {
      __syncthreads();
      for (unsigned base = 0; base < (unsigned)kCand; base += kThreads) {
        const unsigned i = base + t;
        const unsigned p = i ^ jj;
        if (p > i) {
          const unsigned long long a = cand[i], c = cand[p];
          const bool up = ((i & kk) == 0);
          if (up ? (a < c) : (a > c)) { cand[i] = c; cand[p] = a; }
        }
      }
    }
  }
  __syncthreads();

  // ---------- Phase 5: gather + decode top-200 boxes into LDS ----------
  if (t < (unsigned)kKP) {
    float x1 = 0.f, y1 = 0.f, x2 = 0.f, y2 = 0.f, ar = 0.f;
    if (t < (unsigned)kK) {
      const unsigned idx = ~(unsigned)(cand[t] & 0xFFFFFFFFull);
      const float4 rb = reinterpret_cast<const float4*>(boxes_raw)[(size_t)b * kN + idx];
      const float w = rb.z * 0.5f + 1e-2f;
      const float h = rb.w * 0.5f + 1e-2f;
      const float hw = 0.5f * w, hh = 0.5f * h;
      x1 = rb.x - hw; y1 = rb.y - hh; x2 = rb.x + hw; y2 = rb.y + hh;
      ar = w * h;
    }
    bx1[t] = x1; by1[t] = y1; bx2[t] = x2; by2[t] = y2; barea[t] = ar;
    sup[t] = 0u;
  }
  __syncthreads();

  // ---------- Phase 6: Fast-NMS over 16x16 tiles, one wave per tile ----------
  // Suppression predicate (division-free):
  //   iou(i,j) > thr  <=>  inter > thr * ((area_i + area_j) - inter)   [union > 0]
  // The rank-2 area-sum tile S[i][j] = area_i + area_j is produced by
  // V_WMMA_F32_16X16X4_F32 with A row m = [area_m, 1, 0, 0], B col n = [1, area_n, 0, 0].
  const unsigned lane = t & 31u;
  const unsigned wid  = t >> 5;
  const bool     lohalf = lane < 16u;
  const unsigned nl   = lane & 15u;

  for (unsigned tp = wid; tp < (unsigned)kPairs; tp += (kThreads / 32)) {
    unsigned ti = 0, rem = tp;
    while (rem >= (unsigned)kNT - ti) { rem -= (unsigned)kNT - ti; ++ti; }
    const unsigned tj = ti + rem;
    const unsigned i0 = ti * 16u, j0 = tj * 16u;
    const unsigned j  = j0 + nl;                 // this lane's column (C/D layout)

    float usum[8];
#if defined(__HIP_DEVICE_COMPILE__) && __has_builtin(__builtin_amdgcn_wmma_f32_16x16x4_f32)
    // A 16x4 layout: lanes 0-15 hold M=lane, K=0..1 in v0..v1; lanes 16-31 hold K=2..3 (zeros).
    // B 4x16 layout: lanes 0-15 hold N=lane, K=0..1 in v0..v1; lanes 16-31 hold K=2..3 (zeros).
    v2f afrag = { lohalf ? barea[i0 + nl] : 0.0f, lohalf ? 1.0f : 0.0f };
    v2f bfrag = { lohalf ? 1.0f : 0.0f, lohalf ? barea[j0 + nl] : 0.0f };
    v8f acc = {};
    acc = __builtin_amdgcn_wmma_f32_16x16x4_f32(false, afrag, false, bfrag,
                                                (short)0, acc, false, false);
#pragma unroll
    for (int r = 0; r < 8; ++r) usum[r] = acc[r];
#else
#pragma unroll
    for (int r = 0; r < 8; ++r) {
      const unsigned i = i0 + (unsigned)r + (lohalf ? 0u : 8u);
      usum[r] = barea[i] + barea[j];
    }
#endif
    const float xj1 = bx1[j], yj1 = by1[j], xj2 = bx2[j], yj2 = by2[j];
    bool any = false;
#pragma unroll
    for (int r = 0; r < 8; ++r) {
      const unsigned i = i0 + (unsigned)r + (lohalf ? 0u : 8u);  // C/D row for this VGPR
      float iw = fminf(bx2[i], xj2) - fmaxf(bx1[i], xj1);
      float ih = fminf(by2[i], yj2) - fmaxf(by1[i], yj1);
      iw = fmaxf(iw, 0.0f); ih = fmaxf(ih, 0.0f);
      const float inter = iw * ih;
      const bool valid = (i < j) && (i < (unsigned)kK) && (j < (unsigned)kK);
      // iou > thr  <=>  inter > thr*(union)
      any |= valid && (inter > kIouThr * (usum[r] - inter));
    }
    if (j < (unsigned)kK && any)
      atomicOr(&sup[j], 1u);
  }
  __syncthreads();

  // ---------- Phase 7: suppress + pack [score, x1, y1, x2, y2] ----------
  if (t < (unsigned)kK) {
    const unsigned long long e = cand[t];
    const float sc = __uint_as_float((unsigned)(e >> 32));
    const float so = (sup[t] == 0u && sc > kConfThr) ? sc : 0.0f;
    const size_t o = ((size_t)bc * kK + t) * 5u;
    out[o + 0] = so;
    out[o + 1] = bx1[t];
    out[o + 2] = by1[t];
    out[o + 3] = bx2[t];
    out[o + 4] = by2[t];
  }
}

extern "C" void kernel_launch(void* const* d_in, const int* in_sizes, int n_in,
                              void* d_out, int out_size, void* d_ws, size_t ws_size,
                              hipStream_t stream) {
  (void)in_sizes; (void)n_in; (void)d_ws; (void)ws_size; (void)out_size;
  const float* boxes_raw = (const float*)d_in[0];   // [B,N,4]
  const float* scores    = (const float*)d_in[1];   // [B,C,N]
  float* out             = (float*)d_out;           // [B,C,K,5]

  const dim3 grid(kB * kC);
  const dim3 block(kThreads);
  const size_t shmem = (size_t)kN * sizeof(unsigned int);  // 76992 B score-row cache
  yolact_fastnms_fused<<<grid, block, shmem, stream>>>(boxes_raw, scores, out);
}